// SubgraphingLayer_13022340841696
// MI455X (gfx1250) — compile-verified
//
#include <hip/hip_runtime.h>
#include <stdint.h>

// Problem constants (match reference: B, N, F, K = 32, 512, 128, 32)
#define B_  32
#define N_  512
#define F_  128
#define K_  32
#define NW  16          // N/32 bitmask words per adjacency row
#define QSTRIDE 33      // padded LDS queue stride (odd -> conflict-free on 64 banks)

static __device__ __forceinline__ unsigned ballot32(bool p) {
#if __has_builtin(__builtin_amdgcn_ballot_w32)
    return __builtin_amdgcn_ballot_w32(p);
#else
    return (unsigned)__ballot(p);
#endif
}

// ---------------------------------------------------------------------------
// Kernel A: pack adjacency rows (512 floats) into 512-bit masks (16 u32).
// One wave32 per row; 16 ballots; coalesced 64B store by lanes 0..15.
// Adjacency floats are read exactly once ever -> non-temporal loads so the
// 33.5 MB doesn't evict the L2-resident edges/features working set.
// ---------------------------------------------------------------------------
__global__ void pack_adj_kernel(const float* __restrict__ adj,
                                unsigned* __restrict__ bits) {
    const int lane = threadIdx.x & 31;
    const int wave = threadIdx.x >> 5;
    const long row = (long)blockIdx.x * 8 + wave;      // 0 .. B*N-1
    const float* arow = adj + row * N_;
    unsigned myword = 0u;
#pragma unroll
    for (int w = 0; w < NW; ++w) {
        float v = __builtin_nontemporal_load(&arow[w * 32 + lane]);
        unsigned m = ballot32(v > 0.0f);               // bit 'lane' = col w*32+lane
        if (lane == w) myword = m;
    }
    if (lane < NW) bits[row * NW + lane] = myword;
}

// ---------------------------------------------------------------------------
// Kernel B: one thread per (b, v). FIFO BFS over bitmask adjacency.
// queue[:K] is frozen once tail >= K (later appends land past K), so we exit
// early. Then insertion-sort the K indices (matches jnp.sort ascending).
// ---------------------------------------------------------------------------
__global__ void bfs_kernel(const unsigned* __restrict__ bits,
                           int* __restrict__ idx) {
    __shared__ int qs[256 * QSTRIDE];
    int* q = &qs[threadIdx.x * QSTRIDE];

    const int gid = blockIdx.x * 256 + threadIdx.x;    // 0 .. B*N-1
    const int b   = gid >> 9;                          // / N_
    const int v   = gid & (N_ - 1);
    const unsigned* base = bits + (size_t)b * N_ * NW;

    unsigned disc[NW];
#pragma unroll
    for (int w = 0; w < NW; ++w) disc[w] = 0u;
    disc[v >> 5] |= 1u << (v & 31);

    q[0] = v;
    int tail = 1;

    for (int step = 0; step < K_; ++step) {
        if (tail >= K_) break;
        const int node = (step < tail) ? q[step] : v;  // guard (graph is connected)
        const unsigned* row = base + (size_t)node * NW;
        for (int w = 0; w < NW; ++w) {
            unsigned m = row[w] & ~disc[w];            // new neighbors, ascending index
            disc[w] |= m;
            while (m) {
                int bit = __ffs(m) - 1;
                m &= m - 1u;
                q[tail++] = (w << 5) | bit;
                if (tail >= K_) break;
            }
            if (tail >= K_) break;
        }
    }
    for (int t2 = tail; t2 < K_; ++t2) q[t2] = N_ - 1; // pad (never hit: connected graph)

    // insertion sort, 32 elements
    for (int i = 1; i < K_; ++i) {
        int key = q[i];
        int j = i - 1;
        while (j >= 0 && q[j] > key) { q[j + 1] = q[j]; --j; }
        q[j + 1] = key;
    }

    int* out = idx + (size_t)gid * K_;
#pragma unroll
    for (int i = 0; i < K_; ++i) out[i] = q[i];
}

// ---------------------------------------------------------------------------
// Kernel C: one block (256 thr = 8 waves) per (b, v).
//   wf : global -> LDS -> global via CDNA5 async DMA (ASYNCcnt path); the
//        DMA loads are issued FIRST and stay in flight while the wa/we work
//        runs, then the wave drains ASYNCcnt and fires the async stores
//        (non-temporal: output is write-once, keep L2 for the gather set).
//   wa : rebuilt from adjacency bitmask staged in LDS (no global gather).
//   we : 32x32 scattered gather (rows are L2-resident).
// ---------------------------------------------------------------------------
__global__ void gather_kernel(const float* __restrict__ edges,
                              const float* __restrict__ feat,
                              const unsigned* __restrict__ bits,
                              const int* __restrict__ idx,
                              float* __restrict__ wa,
                              float* __restrict__ we,
                              float* __restrict__ wf) {
    __shared__ int six[K_];
    __shared__ unsigned lbits[K_ * NW];
    __shared__ __align__(16) float lbuf[K_ * F_];      // 16 KB staging

    const int t    = threadIdx.x;
    const int lane = t & 31;
    const int wave = t >> 5;
    const int gid  = blockIdx.x;                       // (b, v) flat
    const int b    = gid >> 9;

    if (t < K_) {
        int r = idx[(size_t)gid * K_ + t];
        six[t] = (r < N_) ? r : (N_ - 1);              // clamp (JAX clamps OOB)
    }
    __syncthreads();

    // ---- issue async DMA loads: 4 feature rows per wave, 16 B per lane ----
#pragma unroll
    for (int j = 0; j < 4; ++j) {
        const int r = wave * 4 + j;
        const float* g = feat + ((size_t)b * N_ + six[r]) * F_ + lane * 4;
        unsigned           ldsa = (unsigned)(uintptr_t)(&lbuf[r * F_ + lane * 4]);
        unsigned long long ga   = (unsigned long long)(uintptr_t)g;
        asm volatile("global_load_async_to_lds_b128 %0, %1, off"
                     :: "v"(ldsa), "v"(ga) : "memory");
    }

    // ---- while DMA is in flight: stage 32 bitmask rows (2 KB) into LDS ----
    {
        const unsigned* bbase = bits + (size_t)b * N_ * NW;
        for (int e = t; e < K_ * NW; e += 256) {
            int r = e >> 4, w = e & (NW - 1);
            lbits[e] = bbase[(size_t)six[r] * NW + w];
        }
    }
    __syncthreads();                                   // barrier does not drain ASYNCcnt

    // ---- wa from bitmask, we scattered gather (still overlapping DMA) ----
    float*       wab   = wa + (size_t)gid * K_ * K_;
    float*       web   = we + (size_t)gid * K_ * K_;
    const float* ebase = edges + (size_t)b * N_ * N_;
#pragma unroll
    for (int i = 0; i < 4; ++i) {
        const int e = t + 256 * i;                     // 0..1023
        const int r = e >> 5, c = e & 31;
        const int cc = six[c];
        const unsigned bit = (lbits[r * NW + (cc >> 5)] >> (cc & 31)) & 1u;
        __builtin_nontemporal_store(bit ? 1.0f : 0.0f, &wab[e]);
        __builtin_nontemporal_store(ebase[(size_t)six[r] * N_ + cc], &web[e]);
    }

    // ---- drain DMA loads, then stream wf out via async stores (NT) ----
    asm volatile("s_wait_asynccnt 0" ::: "memory");
    float* wfb = wf + (size_t)gid * K_ * F_;
#pragma unroll
    for (int j = 0; j < 4; ++j) {
        const int r = wave * 4 + j;
        float* o = wfb + r * F_ + lane * 4;
        unsigned           ldsa = (unsigned)(uintptr_t)(&lbuf[r * F_ + lane * 4]);
        unsigned long long ga   = (unsigned long long)(uintptr_t)o;
        asm volatile("global_store_async_from_lds_b128 %0, %1, off th:TH_STORE_NT"
                     :: "v"(ga), "v"(ldsa) : "memory");
    }
    asm volatile("s_wait_asynccnt 0" ::: "memory");    // drain before wave exit
}

// ---------------------------------------------------------------------------
extern "C" void kernel_launch(void* const* d_in, const int* in_sizes, int n_in,
                              void* d_out, int out_size, void* d_ws, size_t ws_size,
                              hipStream_t stream) {
    const float* feat  = (const float*)d_in[0];        // (B, N, F)
    const float* adj   = (const float*)d_in[1];        // (B, N, N)
    const float* edges = (const float*)d_in[2];        // (B, N, N)
    // d_in[3] = neigh_size (== K_, compile-time)

    // workspace: [ bitmask adjacency : 1 MB ][ sorted BFS indices : 2 MB ]
    unsigned* bits = (unsigned*)d_ws;
    int*      idx  = (int*)((char*)d_ws + (size_t)B_ * N_ * NW * sizeof(unsigned));

    float* wa = (float*)d_out;                         // (B, N, K, K)
    float* we = wa + (size_t)B_ * N_ * K_ * K_;        // (B, N, K, K)
    float* wf = we + (size_t)B_ * N_ * K_ * K_;        // (B, N, K, F)

    pack_adj_kernel<<<(B_ * N_) / 8, 256, 0, stream>>>(adj, bits);
    bfs_kernel    <<<(B_ * N_) / 256, 256, 0, stream>>>(bits, idx);
    gather_kernel <<<B_ * N_, 256, 0, stream>>>(edges, feat, bits, idx, wa, we, wf);
}